// PokerFusedEmbedding_58712202936643
// MI455X (gfx1250) — compile-verified
//
#include <hip/hip_runtime.h>
#include <hip/hip_bf16.h>
#include <math.h>

// ---------------------------------------------------------------------------
// Poker fused embedding for MI455X (gfx1250, wave32).
// One fused kernel: per 16-token tile, run the two masked MLP GEMMs with
// V_WMMA_F32_16X16X4_F32 (exact f32 matrix path), LayerNorm+ReLU in LDS,
// then assemble all embedding terms and store once. Memory-bound (~160MB
// traffic -> ~7us at 23.3TB/s), so everything stays fused; no HBM temps.
// ---------------------------------------------------------------------------

typedef __attribute__((ext_vector_type(2))) float v2f;
typedef __attribute__((ext_vector_type(8))) float v8f;

#define D_DIM      256
#define SEQ        512
#define NBINS      16
#define CTXK       143
#define CTXK_PAD   144
#define TM         16
#define PAD_ID     72
#define CARD_OFF_C 4
#define ACT_OFF_C  56
#define LN_EPS_F   1e-5f

__device__ __forceinline__ int iclamp(int v, int lo, int hi) {
    return v < lo ? lo : (v > hi ? hi : v);
}

union V8U { v8f v; float f[8]; };

// LayerNorm (+bias already added) + ReLU over a 16 x 256 LDS tile.
// Two-stage reduction: 256 threads -> (row,seg) partials -> 16 threads finish.
__device__ __forceinline__ void tile_layernorm_relu(
    float (*tile)[D_DIM], const float* __restrict__ g, const float* __restrict__ beta,
    float (*red)[TM][16], float* mu_s, float* rstd_s, int tid)
{
    int row = tid >> 4;
    int seg = tid & 15;
    float s = 0.f, q = 0.f;
    #pragma unroll
    for (int i = 0; i < 16; ++i) {
        float v = tile[row][seg * 16 + i];
        s += v; q += v * v;
    }
    red[0][row][seg] = s;
    red[1][row][seg] = q;
    __syncthreads();
    if (tid < TM) {
        float ts = 0.f, tq = 0.f;
        #pragma unroll
        for (int i = 0; i < 16; ++i) { ts += red[0][tid][i]; tq += red[1][tid][i]; }
        float mu  = ts * (1.f / D_DIM);
        float var = tq * (1.f / D_DIM) - mu * mu;
        mu_s[tid]   = mu;
        rstd_s[tid] = rsqrtf(var + LN_EPS_F);
    }
    __syncthreads();
    float mu = mu_s[row], rstd = rstd_s[row];
    #pragma unroll
    for (int i = 0; i < 16; ++i) {
        int d = seg * 16 + i;
        float v = (tile[row][d] - mu) * rstd * g[d] + beta[d];
        tile[row][d] = fmaxf(v, 0.f);
    }
    __syncthreads();
}

__global__ __launch_bounds__(256) void poker_embed_kernel(
    const int*   __restrict__ token_ids,
    const int*   __restrict__ token_streets,
    const int*   __restrict__ card_ranks,
    const int*   __restrict__ card_suits,
    const int*   __restrict__ action_actors,
    const float* __restrict__ legal_masks,   // (N*S, 16)
    const float* __restrict__ ctx_feats,     // (N*S, 9)
    const float* __restrict__ base_emb,      // (73, D)
    const float* __restrict__ street_emb,    // (4, D)
    const float* __restrict__ rank_emb,      // (13, D)
    const float* __restrict__ suit_emb,      // (4, D)
    const float* __restrict__ actor_emb,     // (2, D)
    const float* __restrict__ atype_emb,     // (16, D)
    const float* __restrict__ legal_W,  const float* __restrict__ legal_b,
    const float* __restrict__ legal_g,  const float* __restrict__ legal_beta,
    const float* __restrict__ game_W,   const float* __restrict__ game_b,
    const float* __restrict__ game_g,   const float* __restrict__ game_beta,
    const float* __restrict__ ctx_W,    const float* __restrict__ ctx_b,
    const float* __restrict__ ctx_g,    const float* __restrict__ ctx_beta,
    float* __restrict__ out)
{
    __shared__ float sLegal[TM][D_DIM];      // legal MLP tile result
    __shared__ float sCtx[TM][D_DIM];        // ctx MLP tile result
    __shared__ float sA[TM][CTXK_PAD];       // ctx fourier features (A matrix)
    __shared__ float sAleg[TM][NBINS];       // legal mask features (A matrix)
    __shared__ float sGame[D_DIM];           // game MLP result (one row)
    __shared__ float sRed[2][TM][16];        // LN reduction partials
    __shared__ float sMu[TM], sRstd[TM];
    __shared__ float sGMu, sGRstd;
    __shared__ int sIds[TM], sPadF[TM], sStreet[TM], sRank[TM], sSuit[TM], sActor[TM];
    __shared__ int sActFlag[TM], sCtxFlag[TM];

    const int tid  = threadIdx.x;
    const int lane = tid & 31;
    const int wave = tid >> 5;
    const int t0   = blockIdx.x * TM;        // first token of tile
    const int bIdx = t0 / SEQ;               // batch row (tiles never cross rows)
    const bool isGameTile = (t0 % SEQ) == 0;

    // per-batch scalars (broadcast loads)
    const float sbv   = ctx_feats[(size_t)bIdx * SEQ * 9 + 0];
    const float bbv   = ctx_feats[(size_t)bIdx * SEQ * 9 + 1];
    const float scale = 100.f * bbv;
    const float ssafe = (scale == 0.f) ? 1.f : scale;
    const float bbsafe = (bbv == 0.f) ? 1.f : bbv;

    // ---- per-token metadata ----
    if (tid < TM) {
        int t  = t0 + tid;
        int id = token_ids[t];
        int p  = (id < 0) ? 1 : 0;
        if (p) id = PAD_ID;
        sIds[tid]    = id;
        sPadF[tid]   = p;
        sStreet[tid] = iclamp(token_streets[t], 0, 3);
        sRank[tid]   = iclamp(card_ranks[t], 0, 12);
        sSuit[tid]   = iclamp(card_suits[t], 0, 3);
        sActor[tid]  = iclamp(action_actors[t], 0, 1);
        sActFlag[tid] = (id >= ACT_OFF_C) && (id < ACT_OFF_C + NBINS);
        sCtxFlag[tid] = (id == 1);
    }
    __syncthreads();

    int anyAct = 0, anyCtx = 0;
    #pragma unroll
    for (int i = 0; i < TM; ++i) { anyAct |= sActFlag[i]; anyCtx |= sCtxFlag[i]; }
    // anyAct/anyCtx are uniform across the block -> EXEC stays all-ones inside.

    const int m16 = lane & 15;   // A: row M, B: col N (within 16-tile)
    const int kh  = lane >> 4;   // lane-half selects K pair (K+0/1 vs K+2/3)

    // ================= legal MLP: (16x16) @ (16x256), K=16 =================
    if (anyAct) {
        { // stage A matrix into LDS
            int m = tid >> 4, k = tid & 15;
            sAleg[m][k] = legal_masks[(size_t)(t0 + m) * NBINS + k];
        }
        __syncthreads();

        v8f c0 = {0.f,0.f,0.f,0.f,0.f,0.f,0.f,0.f};
        v8f c1 = {0.f,0.f,0.f,0.f,0.f,0.f,0.f,0.f};
        const int n0 = wave * 32 + m16;      // first N tile column
        const int n1 = n0 + 16;              // second N tile column
        #pragma unroll
        for (int k0 = 0; k0 < NBINS; k0 += 4) {
            int k = k0 + 2 * kh;
            v2f a;  a.x  = sAleg[m16][k];            a.y  = sAleg[m16][k + 1];
            v2f b0; b0.x = legal_W[n0 * NBINS + k];  b0.y = legal_W[n0 * NBINS + k + 1];
            v2f b1; b1.x = legal_W[n1 * NBINS + k];  b1.y = legal_W[n1 * NBINS + k + 1];
            c0 = __builtin_amdgcn_wmma_f32_16x16x4_f32(false, a, false, b0, (short)0, c0, false, false);
            c1 = __builtin_amdgcn_wmma_f32_16x16x4_f32(false, a, false, b1, (short)0, c1, false, false);
        }
        V8U u0; u0.v = c0;
        V8U u1; u1.v = c1;
        const int mb = kh * 8;
        const float bA = legal_b[n0], bB = legal_b[n1];
        #pragma unroll
        for (int r = 0; r < 8; ++r) {
            sLegal[mb + r][n0] = u0.f[r] + bA;
            sLegal[mb + r][n1] = u1.f[r] + bB;
        }
        __syncthreads();
        tile_layernorm_relu(sLegal, legal_g, legal_beta, sRed, sMu, sRstd, tid);
    }

    // ================= ctx MLP: (16x143) @ (143x256), K pad to 144 =========
    if (anyCtx) {
        if (tid < TM) sA[tid][CTXK] = 0.f;   // K padding column
        if (tid < TM * 13) {
            const int m = tid / 13, j = tid % 13;
            const float* cf = ctx_feats + (size_t)(t0 + m) * 9;
            const float pot = cf[0];
            const float potsafe = (pot == 0.f) ? 1.f : pot;
            float pj;
            switch (j) {
                case 0:  pj = pot   / ssafe;   break;
                case 1:  pj = cf[1] / ssafe;   break;
                case 2:  pj = cf[2] / ssafe;   break;
                case 3:  pj = cf[3] / ssafe;   break;
                case 4:  pj = cf[4] / ssafe;   break;
                case 5:  pj = cf[5];           break;
                case 6:  pj = cf[6];           break;
                case 7:  pj = cf[7] / ssafe;   break;
                case 8:  pj = cf[8] / ssafe;   break;
                case 9:  pj = cf[1] / bbsafe;  break;
                case 10: pj = cf[2] / bbsafe;  break;
                case 11: pj = cf[1] / potsafe; break;
                default: pj = cf[2] / potsafe; break;
            }
            const float FOUR[5] = {3.14159265358979f, 6.28318530717959f,
                                   12.5663706143592f, 25.1327412287183f,
                                   50.2654824574367f};
            float* dst = &sA[m][j * 11];
            dst[0] = pj;
            #pragma unroll
            for (int i = 0; i < 5; ++i) {
                float f = pj * FOUR[i];
                dst[1 + i] = sinf(f);
                dst[6 + i] = cosf(f);
            }
        }
        __syncthreads();

        v8f c0 = {0.f,0.f,0.f,0.f,0.f,0.f,0.f,0.f};
        v8f c1 = {0.f,0.f,0.f,0.f,0.f,0.f,0.f,0.f};
        const int n0 = wave * 32 + m16;
        const int n1 = n0 + 16;
        #pragma unroll 4
        for (int k0 = 0; k0 < CTXK_PAD; k0 += 4) {
            int k = k0 + 2 * kh;           // max k = 142; sA has zero col 143
            v2f a; a.x = sA[m16][k]; a.y = sA[m16][k + 1];
            const float* w0 = ctx_W + (size_t)n0 * CTXK + k;
            const float* w1 = ctx_W + (size_t)n1 * CTXK + k;
            const bool hasHi = (k + 1) < CTXK;
            v2f b0; b0.x = w0[0]; b0.y = hasHi ? w0[1] : 0.f;
            v2f b1; b1.x = w1[0]; b1.y = hasHi ? w1[1] : 0.f;
            c0 = __builtin_amdgcn_wmma_f32_16x16x4_f32(false, a, false, b0, (short)0, c0, false, false);
            c1 = __builtin_amdgcn_wmma_f32_16x16x4_f32(false, a, false, b1, (short)0, c1, false, false);
        }
        V8U u0; u0.v = c0;
        V8U u1; u1.v = c1;
        const int mb = kh * 8;
        const float bA = ctx_b[n0], bB = ctx_b[n1];
        #pragma unroll
        for (int r = 0; r < 8; ++r) {
            sCtx[mb + r][n0] = u0.f[r] + bA;
            sCtx[mb + r][n1] = u1.f[r] + bB;
        }
        __syncthreads();
        tile_layernorm_relu(sCtx, ctx_g, ctx_beta, sRed, sMu, sRstd, tid);
    }

    // ================= game MLP (tiles containing s==0 only) ================
    if (isGameTile) {
        const float pos = ctx_feats[(size_t)bIdx * SEQ * 9 + 2];
        const float ssg = (scale == 0.f) ? 1e-8f : scale;
        const float gf[5] = { sbv, bbv, pos, bbv / ssg, sbv / ssg };
        const float* w = game_W + tid * 5;
        float h = game_b[tid];
        #pragma unroll
        for (int k = 0; k < 5; ++k) h += gf[k] * w[k];
        sGame[tid] = h;
        __syncthreads();
        if (tid < 16) {
            float s = 0.f, q = 0.f;
            #pragma unroll
            for (int i = 0; i < 16; ++i) {
                float v = sGame[tid * 16 + i];
                s += v; q += v * v;
            }
            sRed[0][0][tid] = s; sRed[1][0][tid] = q;
        }
        __syncthreads();
        if (tid == 0) {
            float s = 0.f, q = 0.f;
            #pragma unroll
            for (int i = 0; i < 16; ++i) { s += sRed[0][0][i]; q += sRed[1][0][i]; }
            float mu  = s * (1.f / D_DIM);
            float var = q * (1.f / D_DIM) - mu * mu;
            sGMu = mu; sGRstd = rsqrtf(var + LN_EPS_F);
        }
        __syncthreads();
        float v = (h - sGMu) * sGRstd * game_g[tid] + game_beta[tid];
        sGame[tid] = fmaxf(v, 0.f);
    }
    __syncthreads();

    // ================= assembly + single global store =======================
    {
        const int m = tid >> 4;
        const int t = t0 + m;
        const int id     = sIds[m];
        const int padF   = sPadF[m];
        const bool cardm = (id >= CARD_OFF_C) && (id < CARD_OFF_C + 52);
        const bool actm  = (id >= ACT_OFF_C)  && (id < ACT_OFF_C + NBINS);
        const bool ctxm  = (id == 1);
        const bool addGame = isGameTile && (m == 0);
        const int aid = iclamp(id - ACT_OFF_C, 0, NBINS - 1);
        const float* be = base_emb   + (size_t)id * D_DIM;
        const float* se = street_emb + (size_t)sStreet[m] * D_DIM;
        const float* re = rank_emb   + (size_t)sRank[m]   * D_DIM;
        const float* su = suit_emb   + (size_t)sSuit[m]   * D_DIM;
        const float* ae = actor_emb  + (size_t)sActor[m]  * D_DIM;
        const float* te = atype_emb  + (size_t)aid * D_DIM;
        float* op = out + (size_t)t * D_DIM;
        const int dbase = (tid & 15) * 16;
        #pragma unroll
        for (int i = 0; i < 16; ++i) {
            int d = dbase + i;
            float v = be[d] + se[d];
            if (cardm)   v += re[d] + su[d];
            if (actm)    v += ae[d] + te[d] + sLegal[m][d];
            if (addGame) v += sGame[d];
            if (ctxm)    v += sCtx[m][d];
            if (padF)    v = 0.f;
            op[d] = v;
        }
    }
}

extern "C" void kernel_launch(void* const* d_in, const int* in_sizes, int n_in,
                              void* d_out, int out_size, void* d_ws, size_t ws_size,
                              hipStream_t stream) {
    const int*   token_ids     = (const int*)d_in[0];
    const int*   token_streets = (const int*)d_in[1];
    const int*   card_ranks    = (const int*)d_in[2];
    const int*   card_suits    = (const int*)d_in[3];
    const int*   action_actors = (const int*)d_in[4];
    const float* legal_masks   = (const float*)d_in[5];
    const float* ctx_feats     = (const float*)d_in[6];
    const float* base_emb      = (const float*)d_in[7];
    const float* street_emb    = (const float*)d_in[8];
    const float* rank_emb      = (const float*)d_in[9];
    const float* suit_emb      = (const float*)d_in[10];
    const float* actor_emb     = (const float*)d_in[11];
    const float* atype_emb     = (const float*)d_in[12];
    const float* legal_W       = (const float*)d_in[13];
    const float* legal_b       = (const float*)d_in[14];
    const float* legal_g       = (const float*)d_in[15];
    const float* legal_beta    = (const float*)d_in[16];
    const float* game_W        = (const float*)d_in[17];
    const float* game_b        = (const float*)d_in[18];
    const float* game_g        = (const float*)d_in[19];
    const float* game_beta     = (const float*)d_in[20];
    const float* ctx_W         = (const float*)d_in[21];
    const float* ctx_b         = (const float*)d_in[22];
    const float* ctx_g         = (const float*)d_in[23];
    const float* ctx_beta      = (const float*)d_in[24];
    float* out = (float*)d_out;

    const int ntok    = in_sizes[0];       // N * S
    const int nblocks = ntok / TM;

    poker_embed_kernel<<<dim3(nblocks), dim3(256), 0, stream>>>(
        token_ids, token_streets, card_ranks, card_suits, action_actors,
        legal_masks, ctx_feats, base_emb, street_emb, rank_emb, suit_emb,
        actor_emb, atype_emb,
        legal_W, legal_b, legal_g, legal_beta,
        game_W, game_b, game_g, game_beta,
        ctx_W, ctx_b, ctx_g, ctx_beta,
        out);
}